// BitInput_64991445123116
// MI455X (gfx1250) — compile-verified
//
#include <hip/hip_runtime.h>
#include <stdint.h>

// Per-row selection of the k smallest of 512 uniforms, k = round(p*512).
// One wave32 per row; async global->LDS double-buffered staging (CDNA5 path).

#define NBITS 512
#define WAVES_PER_BLOCK 8
#define CAND_MAX 64

typedef float vf4 __attribute__((ext_vector_type(4)));

__global__ __launch_bounds__(256, 1) void bitstream_select_kernel(
    const float* __restrict__ p,
    const float* __restrict__ u,
    float* __restrict__ out,
    int R)
{
    // Per-wave LDS working set (~5.8 KB * 8 waves ~= 46 KB)
    __shared__ uint4    stage[WAVES_PER_BLOCK][2][NBITS / 4]; // 2KB per buffer
    __shared__ uint4    hist4[WAVES_PER_BLOCK][64];           // 256 bins (u32)
    __shared__ uint4    scanb4[WAVES_PER_BLOCK][8];           // 32 lane totals
    __shared__ unsigned sel  [WAVES_PER_BLOCK][2];
    __shared__ unsigned cand [WAVES_PER_BLOCK][2 * CAND_MAX];
    __shared__ unsigned ccnt [WAVES_PER_BLOCK];

    const int lane   = threadIdx.x & 31;
    const int w      = threadIdx.x >> 5;
    const int waveId = blockIdx.x * WAVES_PER_BLOCK + w;
    const int stride = gridDim.x * WAVES_PER_BLOCK;

    unsigned* const hist = (unsigned*)&hist4[w][0];

    // LDS byte offsets of this wave's two stage buffers (low 32 bits of the
    // generic LDS pointer are the LDS byte address on gfx1250).
    const unsigned ldsBase0 = (unsigned)(uintptr_t)(&stage[w][0][0]);
    const unsigned ldsBase1 = (unsigned)(uintptr_t)(&stage[w][1][0]);

    int row = waveId;
    if (row >= R) return;

    // Prefetch first row into buffer 0: 4 x (32 lanes x 16B) = 2KB async copy.
    {
        const float* src = u + (size_t)row * NBITS;
#pragma unroll
        for (int c = 0; c < 4; ++c) {
            unsigned loff = ldsBase0 + (unsigned)(c * 512 + lane * 16);
            unsigned goff = (unsigned)(c * 512 + lane * 16);
            asm volatile("global_load_async_to_lds_b128 %0, %1, %2"
                         :: "v"(loff), "v"(goff), "s"(src) : "memory");
        }
    }

    int buf = 0;
    for (; row < R; row += stride) {
        const int  nrow     = row + stride;
        const bool havenext = (nrow < R);

        if (havenext) {
            // Issue next row into the other buffer before consuming this one.
            const float*   src = u + (size_t)nrow * NBITS;
            const unsigned nb  = (buf ^ 1) ? ldsBase1 : ldsBase0;
#pragma unroll
            for (int c = 0; c < 4; ++c) {
                unsigned loff = nb + (unsigned)(c * 512 + lane * 16);
                unsigned goff = (unsigned)(c * 512 + lane * 16);
                asm volatile("global_load_async_to_lds_b128 %0, %1, %2"
                             :: "v"(loff), "v"(goff), "s"(src) : "memory");
            }
            // Async loads complete in order: <=4 outstanding => current row done.
            asm volatile("s_wait_asynccnt 0x4" ::: "memory");
        } else {
            asm volatile("s_wait_asynccnt 0x0" ::: "memory");
        }

        // k = round-half-even(p * 512), wave-uniform (scalar load + branch).
        const int kk = (int)rintf(p[row] * 512.0f);
        vf4* const op = (vf4*)(out + (size_t)row * NBITS + lane * 16);

        if (kk <= 0 || kk >= NBITS) {
            const float v = (kk >= NBITS) ? 1.0f : 0.0f;
            vf4 q = {v, v, v, v};
            __builtin_nontemporal_store(q, op + 0);
            __builtin_nontemporal_store(q, op + 1);
            __builtin_nontemporal_store(q, op + 2);
            __builtin_nontemporal_store(q, op + 3);
            buf ^= 1;
            continue;
        }

        // Read this lane's 16 contiguous elements from LDS, build monotone keys.
        const uint4* sp = (buf ? &stage[w][1][0] : &stage[w][0][0]) + lane * 4;
        unsigned key[16];
        {
            uint4 q0 = sp[0], q1 = sp[1], q2 = sp[2], q3 = sp[3];
            unsigned raw[16] = {q0.x, q0.y, q0.z, q0.w, q1.x, q1.y, q1.z, q1.w,
                                q2.x, q2.y, q2.z, q2.w, q3.x, q3.y, q3.z, q3.w};
#pragma unroll
            for (int j = 0; j < 16; ++j)
                key[j] = (unsigned)(__uint_as_float(raw[j]) * 4294967296.0f);
        }

        // Zero histogram (2 x ds_store_b128 per lane) + candidate counter.
        {
            uint4 z = {0u, 0u, 0u, 0u};
            hist4[w][lane * 2 + 0] = z;
            hist4[w][lane * 2 + 1] = z;
        }
        if (lane == 0) ccnt[w] = 0u;
        __asm__ volatile("" ::: "memory");

        // 256-bin histogram of the top key byte (uniform => ~2 elems/bin).
#pragma unroll
        for (int j = 0; j < 16; ++j) atomicAdd(&hist[key[j] >> 24], 1u);
        __asm__ volatile("" ::: "memory");

        // Inclusive prefix over this lane's 8 bins; publish lane total.
        unsigned s[8]; unsigned run = 0;
        {
            uint4 h0 = hist4[w][lane * 2 + 0];
            uint4 h1 = hist4[w][lane * 2 + 1];
            unsigned c[8] = {h0.x, h0.y, h0.z, h0.w, h1.x, h1.y, h1.z, h1.w};
#pragma unroll
            for (int i = 0; i < 8; ++i) { run += c[i]; s[i] = run; }
        }
        ((unsigned*)&scanb4[w][0])[lane] = run;
        __asm__ volatile("" ::: "memory");

        // Cross-lane exclusive sum via 8 x ds_load_b128 + masked adds.
        unsigned excl = 0;
#pragma unroll
        for (int i = 0; i < 8; ++i) {
            uint4 t = scanb4[w][i];
            const int b = i * 4;
            excl += (b + 0 < lane) ? t.x : 0u;
            excl += (b + 1 < lane) ? t.y : 0u;
            excl += (b + 2 < lane) ? t.z : 0u;
            excl += (b + 3 < lane) ? t.w : 0u;
        }

        // Exactly one lane's bin range contains the k-th smallest; it publishes
        // the bucket id B and the count of elements strictly below bucket B.
        const unsigned uk = (unsigned)kk;
        if (excl < uk && excl + run >= uk) {
            int Bj = 7; unsigned below = excl + s[6];
#pragma unroll
            for (int i = 6; i >= 0; --i)
                if (excl + s[i] >= uk) { Bj = i; below = excl + (i ? s[i - 1] : 0u); }
            sel[w][0] = (unsigned)(lane * 8 + Bj);
            sel[w][1] = below;
        }
        __asm__ volatile("" ::: "memory");

        const unsigned B     = sel[w][0];
        const unsigned below = sel[w][1];
        const unsigned need  = uk - below;   // ones still needed inside bucket B

        // Gather the few in-bucket candidates (key, index) into LDS.
        const unsigned e0 = (unsigned)(lane * 16);
#pragma unroll
        for (int j = 0; j < 16; ++j) {
            if ((key[j] >> 24) == B) {
                unsigned slot = atomicAdd(&ccnt[w], 1u);
                if (slot < CAND_MAX) {
                    cand[w][2 * slot]     = key[j];
                    cand[w][2 * slot + 1] = e0 + j;
                }
            }
        }
        __asm__ volatile("" ::: "memory");
        unsigned m = ccnt[w]; if (m > CAND_MAX) m = CAND_MAX;

        // Emit bits: bucket<B -> 1, bucket>B -> 0, bucket==B -> exact rank
        // among candidates (ties broken by index, matching stable argsort).
        float res[16];
#pragma unroll
        for (int j = 0; j < 16; ++j) {
            const unsigned b = key[j] >> 24;
            float bit;
            if (b < B)      bit = 1.0f;
            else if (b > B) bit = 0.0f;
            else {
                unsigned rank = 0;
                const unsigned kj = key[j], ej = e0 + j;
                for (unsigned t = 0; t < m; ++t) {
                    uint2 cd = *(const uint2*)&cand[w][2 * t];  // ds_load_b64
                    rank += (cd.x < kj || (cd.x == kj && cd.y < ej)) ? 1u : 0u;
                }
                bit = (rank < need) ? 1.0f : 0.0f;
            }
            res[j] = bit;
        }

        vf4 r0 = {res[0],  res[1],  res[2],  res[3]};
        vf4 r1 = {res[4],  res[5],  res[6],  res[7]};
        vf4 r2 = {res[8],  res[9],  res[10], res[11]};
        vf4 r3 = {res[12], res[13], res[14], res[15]};
        __builtin_nontemporal_store(r0, op + 0);
        __builtin_nontemporal_store(r1, op + 1);
        __builtin_nontemporal_store(r2, op + 2);
        __builtin_nontemporal_store(r3, op + 3);

        buf ^= 1;
    }
}

extern "C" void kernel_launch(void* const* d_in, const int* in_sizes, int n_in,
                              void* d_out, int out_size, void* d_ws, size_t ws_size,
                              hipStream_t stream) {
    (void)n_in; (void)out_size; (void)d_ws; (void)ws_size;
    const float* p = (const float*)d_in[0];   // [128,1024] float32
    const float* u = (const float*)d_in[1];   // [128,1024,512] float32
    float* out = (float*)d_out;               // [128,1024,512] float32
    const int R = in_sizes[0];                // 131072 rows

    int blocks = (R + 63) / 64;               // 8 waves/block -> ~8 rows per wave
    if (blocks > 2048) blocks = 2048;
    if (blocks < 1)    blocks = 1;

    hipLaunchKernelGGL(bitstream_select_kernel, dim3(blocks), dim3(256), 0, stream,
                       p, u, out, R);
}